// DifferentiableTTFSEncoder_13597866459673
// MI455X (gfx1250) — compile-verified
//
#include <hip/hip_runtime.h>
#include <stdint.h>

typedef __attribute__((ext_vector_type(4))) float        v4f;
typedef __attribute__((ext_vector_type(4))) unsigned int v4u;
typedef __attribute__((ext_vector_type(4))) int          v4i;
typedef __attribute__((ext_vector_type(8))) int          v8i;

#define TTFS_T 64
#define TTFS_N 1024
// exp(-1/TAU) with TAU=2.0, rounded to nearest fp32
#define DECAY_F 0.60653065971263342f

__global__ __launch_bounds__(256) void ttfs_encode_kernel(
    const float* __restrict__ x,
    const float* __restrict__ sens,
    float* __restrict__ out) {
  __shared__ float s_sens[TTFS_N];

  const int b   = (int)blockIdx.x;
  const int tid = (int)threadIdx.x;   // 0..255
  const int n4  = tid << 2;           // this thread's first n

  // ---- Stage the broadcast sensitivity vector (4 KB) into LDS via the
  // ---- Tensor Data Mover: one descriptor, issued by wave 0 only.
#if __has_builtin(__builtin_amdgcn_tensor_load_to_lds)
  if (tid < 32) {
    const unsigned long long ga = (unsigned long long)(uintptr_t)sens;
    // Flat LDS address: low 32 bits of the generic address are the LDS byte
    // address.
    const unsigned lds_addr = (unsigned)(uintptr_t)(&s_sens[0]);

    v4u g0;
    g0.x = 1u;                                     // count=1, user descriptor
    g0.y = lds_addr;                               // lds_addr[31:0]
    g0.z = (unsigned)(ga & 0xffffffffull);         // global_addr[31:0]
    g0.w = (unsigned)((ga >> 32) & 0x1ffffffull)   // global_addr[56:32]
         | (2u << 30);                             // type = 2 ("image")

    v8i g1;
    g1[0] = 0x00020000;                 // workgroup_mask=0, data_size=2 (4B)
    g1[1] = (int)(TTFS_N << 16);        // tensor_dim0[15:0] = 1024 @ bits[79:64... (31:16)]
    g1[2] = (int)(1u << 16);            // tensor_dim0 hi = 0; tensor_dim1 = 1
    g1[3] = (int)(TTFS_N << 16);        // tensor_dim1 hi = 0; tile_dim0 = 1024
    g1[4] = 0;                          // tile_dim1 = 0 (unused), tile_dim2 = 0
    g1[5] = TTFS_N;                     // tensor_dim0_stride = 1024 elements
    g1[6] = 0;                          // stride hi, dim1_stride lo
    g1[7] = 0;

    v4i z4 = {0, 0, 0, 0};
#if defined(__clang_major__) && (__clang_major__ >= 23)
    v8i z8 = {0, 0, 0, 0, 0, 0, 0, 0};
    __builtin_amdgcn_tensor_load_to_lds(g0, g1, z4, z4, z8, 0);
#else
    __builtin_amdgcn_tensor_load_to_lds(g0, g1, z4, z4, 0);
#endif
    __builtin_amdgcn_s_wait_tensorcnt(0);
  }
  __syncthreads();
#else
  for (int i = tid; i < TTFS_N; i += 256) s_sens[i] = sens[i];
  __syncthreads();
#endif

  // ---- Per-thread state: 4 consecutive channels ----
  const float decay = DECAY_F;
  const float omd   = 1.0f - DECAY_F;

  const v4f sv = *(const v4f*)(&s_sens[n4]);
  const v4f xv = *(const v4f*)(x + (size_t)b * TTFS_N + n4);
  v4f cw = (xv * sv) * omd;   // current * (1 - decay), constant per step

  v4f m = {0.0f, 0.0f, 0.0f, 0.0f};
  int fired = 0;              // 4-bit has_fired mask

  float* outp = out + (size_t)b * ((size_t)TTFS_T * TTFS_N) + n4;

  // 64 steps; each step emits one 16B non-temporal store per thread
  // (512 B contiguous per wave per step -> fully coalesced streaming).
#pragma unroll 8
  for (int t = 0; t < TTFS_T; ++t) {
    v4f o;
#pragma unroll
    for (int i = 0; i < 4; ++i) {
      float mi = fmaf(m[i], decay, cw[i]);      // mem = mem*d + c*(1-d)
      const bool sp = (mi >= 1.0f);             // spike = mem >= V_TH
      o[i] = (sp && !((fired >> i) & 1)) ? 1.0f : 0.0f;  // effective spike
      fired |= sp ? (1 << i) : 0;               // has_fired |= spike
      m[i] = sp ? 0.0f : mi;                    // reset on spike
    }
    __builtin_nontemporal_store(o, (v4f*)outp); // th:NT b128 store
    outp += TTFS_N;
  }
}

extern "C" void kernel_launch(void* const* d_in, const int* in_sizes, int n_in,
                              void* d_out, int out_size, void* d_ws, size_t ws_size,
                              hipStream_t stream) {
  (void)n_in; (void)out_size; (void)d_ws; (void)ws_size;
  const float* x    = (const float*)d_in[0];
  const float* sens = (const float*)d_in[1];
  float*       out  = (float*)d_out;

  const int B = in_sizes[0] / TTFS_N;   // 2048
  dim3 grid((unsigned)B), block(256);
  hipLaunchKernelGGL(ttfs_encode_kernel, grid, block, 0, stream, x, sens, out);
}